// DecoderOnlyAttention_28424093565638
// MI455X (gfx1250) — compile-verified
//
#include <hip/hip_runtime.h>
#include <hip/hip_bf16.h>
#include <math.h>

// ---------------------------------------------------------------------------
// DecoderOnlyAttention decode step for MI455X (gfx1250, wave32, WMMA).
// B=16, H=32, KVH=8, HD=128, S=4096, D=4096, Q=1.
// Memory-bound (~1.8 GB traffic @ 23.3 TB/s ≈ 77 us floor) -> stay fp32,
// V_WMMA_F32_16X16X4_F32 for all GEMMs (M=16 == batch), async global->LDS
// double buffering for the V stream in attention, branch-free prefetch.
// ---------------------------------------------------------------------------

typedef float v2f __attribute__((ext_vector_type(2)));
typedef float v8f __attribute__((ext_vector_type(8)));

#define Bn   16
#define Hn   32
#define KVHn 8
#define HDn  128
#define Sn   4096
#define Dn   4096

// ---------------- CDNA5 async global->LDS support (guarded) ----------------
#if __has_builtin(__builtin_amdgcn_global_load_async_to_lds_b128)
#define ASYNC_LDS 1
#else
#define ASYNC_LDS 0
#endif

#if ASYNC_LDS
// Builtin signature (from clang diagnostic): b128 variant takes pointers to
// GCC-vector int4: (global int4*, lds int4*, imm offset, imm cpol).
typedef int v4i_vs __attribute__((vector_size(4 * sizeof(int))));
#define GAS __attribute__((address_space(1)))
#define LAS __attribute__((address_space(3)))

// one b128 (16B) per-lane async copy, tracked by ASYNCcnt.
// Global: generic address == AS1 address numerically.
// LDS: low 32 bits of a generic LDS pointer are the AS3 offset.
__device__ __forceinline__ void async_copy16(const float* g, float* l) {
  __builtin_amdgcn_global_load_async_to_lds_b128(
      (GAS v4i_vs*)(size_t)g,
      (LAS v4i_vs*)(unsigned)(size_t)l, 0, 0);
}
__device__ __forceinline__ void wait_async_le2() {
#if __has_builtin(__builtin_amdgcn_s_wait_asynccnt)
  __builtin_amdgcn_s_wait_asynccnt(2);
#else
  asm volatile("s_wait_asynccnt 0x2" ::: "memory");
#endif
}
__device__ __forceinline__ void wait_async_0() {
#if __has_builtin(__builtin_amdgcn_s_wait_asynccnt)
  __builtin_amdgcn_s_wait_asynccnt(0);
#else
  asm volatile("s_wait_asynccnt 0x0" ::: "memory");
#endif
}
#endif

// ---------------------------------------------------------------------------
// Generic C[16,N] = A[16,K] x B[K,N], fp32, via V_WMMA_F32_16X16X4_F32.
// One wave per 16-column tile; K-loop steps of 4.  Fragment layout per
// CDNA5 ISA 7.12.2 (A: lanes 0-15 K={k0,k0+1}, lanes 16-31 K={k0+2,k0+3};
// B mirrored; C: VGPR r -> M=r / r+8 across lane halves).
// Prefetch uses a clamped (branch-free) index so the unrolled body stays
// straight-line and the scheduler can batch loads across iterations.
// ---------------------------------------------------------------------------
__global__ void gemm16_wmma_f32(const float* __restrict__ A,
                                const float* __restrict__ Bm,
                                float* __restrict__ C, int K, int N) {
  const int lane = threadIdx.x;          // 0..31
  const int n0   = blockIdx.x * 16;
  const int mn   = lane & 15;            // row of A (lanes<16) / col of B
  const int koff = (lane >> 4) << 1;     // 0 or 2

  const float* __restrict__ arow = A + (size_t)mn * K + koff;
  const float* __restrict__ bcol = Bm + (size_t)koff * N + n0 + mn;

  v8f c = {0.f, 0.f, 0.f, 0.f, 0.f, 0.f, 0.f, 0.f};

#pragma unroll 4
  for (int k0 = 0; k0 < K; k0 += 4) {
    // branch-free stream-prefetch of B, 8 iterations ahead
    const int kpf = (k0 + 32 < K) ? (k0 + 32) : k0;
    __builtin_prefetch(bcol + (size_t)kpf * N, 0, 3);
    v2f a, b;
    a.x = arow[k0];
    a.y = arow[k0 + 1];
    b.x = bcol[(size_t)k0 * N];
    b.y = bcol[(size_t)(k0 + 1) * N];
    c = __builtin_amdgcn_wmma_f32_16x16x4_f32(false, a, false, b,
                                              (short)0, c, false, false);
  }

  const int nn = n0 + mn;
  const int mb = (lane >> 4) * 8;
#pragma unroll
  for (int r = 0; r < 8; ++r)
    C[(size_t)(mb + r) * N + nn] = c[r];
}

// ---------------------------------------------------------------------------
// RoPE on q/k in-place in workspace; scatter new k,v rows into the copied
// output caches at current_steps[b].
// ---------------------------------------------------------------------------
__global__ void rope_scatter(float* __restrict__ q_ws, float* __restrict__ k_ws,
                             const float* __restrict__ v_ws,
                             const float* __restrict__ cosp,
                             const float* __restrict__ sinp,
                             const int* __restrict__ steps,
                             float* __restrict__ new_key,
                             float* __restrict__ new_value) {
  const int b = blockIdx.x / 48;
  const int i = blockIdx.x % 48;         // 0..31 q, 32..39 k, 40..47 v
  const int d = threadIdx.x;             // 0..127
  const float c = cosp[b * HDn + d];
  const float s = sinp[b * HDn + d];

  if (i < Hn) {
    float* v = q_ws + ((size_t)b * Hn + i) * HDn;
    float x = v[d];
    float o = v[(d + 64) & 127];
    __syncthreads();                     // all reads before in-place write
    v[d] = x * c + ((d < 64) ? -o : o) * s;
  } else if (i < Hn + KVHn) {
    const int h = i - Hn;
    float* v = k_ws + ((size_t)b * KVHn + h) * HDn;
    float x = v[d];
    float o = v[(d + 64) & 127];
    __syncthreads();
    const float val = x * c + ((d < 64) ? -o : o) * s;
    v[d] = val;
    const int cur = steps[b];
    new_key[(((size_t)b * KVHn + h) * Sn + cur) * HDn + d] = val;
  } else {
    const int h = i - Hn - KVHn;
    const int cur = steps[b];
    new_value[(((size_t)b * KVHn + h) * Sn + cur) * HDn + d] =
        v_ws[((size_t)b * KVHn + h) * HDn + d];
  }
}

// ---------------------------------------------------------------------------
// GQA decode attention: one block per (b, kv_head), 256 threads = 8 waves.
// Pass 1: wave-parallel scores (coalesced K reads + branch-free prefetch,
// wave32 shfl reductions).  Softmax: one wave per group head.  Pass 2: V
// accumulation with async global->LDS double buffering (ASYNCcnt pipeline).
// ---------------------------------------------------------------------------
__global__ void attn_decode(const float* __restrict__ q_ws,
                            const float* __restrict__ k_ws,
                            const float* __restrict__ v_ws,
                            const float* __restrict__ past_key,
                            const float* __restrict__ past_value,
                            const float* __restrict__ mask,
                            const int* __restrict__ steps,
                            float* __restrict__ attn_ws) {
  extern __shared__ float smem[];
  float* sc   = smem;                    // 4 * Sn scores -> probs
  float* comb = smem + 4 * Sn;           // 4 * HDn half-combine
  float* sinv = comb + 4 * HDn;          // 4 reciprocal sums
  float* vbuf = sinv + 4;                // 2 * 16 * HDn async V staging (16B aligned)

  const int b   = blockIdx.x >> 3;
  const int kvh = blockIdx.x & 7;
  const int tid = threadIdx.x;
  const int lane = tid & 31;
  const int w    = tid >> 5;             // wave id 0..7
  const int cur  = steps[b];
  const float scale = 0.08838834764831845f;  // 1/sqrt(128)

  // q fragments: 4 group heads x 4 HD-chunks per lane
  float qr[4][4];
  const float* qbase = q_ws + ((size_t)b * Hn + kvh * 4) * HDn;
#pragma unroll
  for (int h = 0; h < 4; ++h)
#pragma unroll
    for (int j = 0; j < 4; ++j)
      qr[h][j] = qbase[h * HDn + lane + 32 * j];

  const float* kcache = past_key + ((size_t)(b * KVHn + kvh)) * Sn * HDn;
  const float* knew   = k_ws + ((size_t)b * KVHn + kvh) * HDn;
  const float* mrow   = mask + (size_t)b * Sn;

  for (int s = w; s < Sn; s += 8) {
    const int spf = (s + 64 < Sn) ? (s + 64) : s;   // branch-free prefetch
    __builtin_prefetch(kcache + (size_t)spf * HDn + lane, 0, 3);
    const float* kr = (s == cur) ? knew : (kcache + (size_t)s * HDn);
    float p0 = 0.f, p1 = 0.f, p2 = 0.f, p3 = 0.f;
#pragma unroll
    for (int j = 0; j < 4; ++j) {
      const float kv = kr[lane + 32 * j];
      p0 += qr[0][j] * kv; p1 += qr[1][j] * kv;
      p2 += qr[2][j] * kv; p3 += qr[3][j] * kv;
    }
#pragma unroll
    for (int off = 16; off > 0; off >>= 1) {
      p0 += __shfl_xor(p0, off, 32);
      p1 += __shfl_xor(p1, off, 32);
      p2 += __shfl_xor(p2, off, 32);
      p3 += __shfl_xor(p3, off, 32);
    }
    if (lane == 0) {
      const float mv = mrow[s];
      sc[0 * Sn + s] = p0 * scale + mv;
      sc[1 * Sn + s] = p1 * scale + mv;
      sc[2 * Sn + s] = p2 * scale + mv;
      sc[3 * Sn + s] = p3 * scale + mv;
    }
  }
  __syncthreads();

  if (w < 4) {                           // wave w owns group head w
    float* row = sc + w * Sn;
    float mx = -3.4e38f;
    for (int s = lane; s < Sn; s += 32) mx = fmaxf(mx, row[s]);
#pragma unroll
    for (int off = 16; off > 0; off >>= 1)
      mx = fmaxf(mx, __shfl_xor(mx, off, 32));
    float sum = 0.f;
    for (int s = lane; s < Sn; s += 32) {
      const float e = __expf(row[s] - mx);
      row[s] = e;
      sum += e;
    }
#pragma unroll
    for (int off = 16; off > 0; off >>= 1) sum += __shfl_xor(sum, off, 32);
    if (lane == 0) sinv[w] = 1.0f / sum;
  }
  __syncthreads();

  // -------- Pass 2: V accumulation --------
  const int d    = tid & 127;
  const int half = tid >> 7;
  const float* vcache = past_value + ((size_t)(b * KVHn + kvh)) * Sn * HDn;
  const float vnew = v_ws[((size_t)b * KVHn + kvh) * HDn + d];
  float a0 = 0.f, a1 = 0.f, a2 = 0.f, a3 = 0.f;

#if ASYNC_LDS
  // 16-row (8KB) chunks, double-buffered via GLOBAL_LOAD_ASYNC_TO_LDS_B128.
  const int CH = 16;                     // rows per chunk
  const int NC = Sn / CH;                // 256 chunks
  {                                      // prologue: chunks 0 and 1 in flight
    const float* s0 = vcache;
    async_copy16(s0 + tid * 4,         vbuf + tid * 4);
    async_copy16(s0 + (tid + 256) * 4, vbuf + (tid + 256) * 4);
    const float* s1 = vcache + CH * HDn;
    float* b1 = vbuf + CH * HDn;
    async_copy16(s1 + tid * 4,         b1 + tid * 4);
    async_copy16(s1 + (tid + 256) * 4, b1 + (tid + 256) * 4);
  }
  for (int c = 0; c < NC; ++c) {
    if (c + 1 < NC) wait_async_le2();    // chunk c landed (c+1 may be in flight)
    else            wait_async_0();
    __syncthreads();                     // all waves' staging visible
    const float* vb = vbuf + (c & 1) * (CH * HDn);
    const int rbase = half * 8;
    const int sbase = c * CH + rbase;
#pragma unroll
    for (int r = 0; r < 8; ++r) {
      const int s = sbase + r;
      float vv = vb[(rbase + r) * HDn + d];
      if (s == cur) vv = vnew;
      a0 += sc[0 * Sn + s] * vv; a1 += sc[1 * Sn + s] * vv;
      a2 += sc[2 * Sn + s] * vv; a3 += sc[3 * Sn + s] * vv;
    }
    __syncthreads();                     // done reading buf before re-fill
    if (c + 2 < NC) {
      const float* sn = vcache + (size_t)(c + 2) * CH * HDn;
      float* bn = vbuf + (c & 1) * (CH * HDn);
      async_copy16(sn + tid * 4,         bn + tid * 4);
      async_copy16(sn + (tid + 256) * 4, bn + (tid + 256) * 4);
    }
  }
#else
  (void)vbuf;
  const int sbeg = half * (Sn / 2), send = sbeg + (Sn / 2);
#pragma unroll 4
  for (int s = sbeg; s < send; ++s) {
    const int spf = (s + 32 < send) ? (s + 32) : s;
    __builtin_prefetch(vcache + (size_t)spf * HDn + d, 0, 3);
    const float vv = (s == cur) ? vnew : vcache[(size_t)s * HDn + d];
    a0 += sc[0 * Sn + s] * vv; a1 += sc[1 * Sn + s] * vv;
    a2 += sc[2 * Sn + s] * vv; a3 += sc[3 * Sn + s] * vv;
  }
#endif

  if (half == 1) {
    comb[0 * HDn + d] = a0; comb[1 * HDn + d] = a1;
    comb[2 * HDn + d] = a2; comb[3 * HDn + d] = a3;
  }
  __syncthreads();
  if (half == 0) {
    a0 += comb[0 * HDn + d]; a1 += comb[1 * HDn + d];
    a2 += comb[2 * HDn + d]; a3 += comb[3 * HDn + d];
    float* outp = attn_ws + (size_t)b * Dn + (kvh * 4) * HDn;
    outp[0 * HDn + d] = a0 * sinv[0];
    outp[1 * HDn + d] = a1 * sinv[1];
    outp[2 * HDn + d] = a2 * sinv[2];
    outp[3 * HDn + d] = a3 * sinv[3];
  }
}

// ---------------------------------------------------------------------------
extern "C" void kernel_launch(void* const* d_in, const int* in_sizes, int n_in,
                              void* d_out, int out_size, void* d_ws, size_t ws_size,
                              hipStream_t stream) {
  const float* hidden     = (const float*)d_in[0];   // [16,1,4096]
  const float* mask       = (const float*)d_in[1];   // [16,1,1,4096]
  const int*   steps      = (const int*)  d_in[2];   // [16]
  const float* cosp       = (const float*)d_in[3];   // [16,1,1,128]
  const float* sinp       = (const float*)d_in[4];   // [16,1,1,128]
  const float* past_key   = (const float*)d_in[5];   // [16,8,4096,128]
  const float* past_value = (const float*)d_in[6];
  const float* Wq         = (const float*)d_in[7];   // [4096,4096]
  const float* Wk         = (const float*)d_in[8];   // [4096,1024]
  const float* Wv         = (const float*)d_in[9];   // [4096,1024]
  const float* Wo         = (const float*)d_in[10];  // [4096,4096]

  float* out       = (float*)d_out;
  float* out_attn  = out;                                   // 16*4096
  float* out_key   = out + (size_t)Bn * Dn;                 // 16*8*4096*128
  float* out_value = out_key + (size_t)Bn * KVHn * Sn * HDn;

  float* ws   = (float*)d_ws;
  float* q_ws = ws;                                  // 16*4096
  float* k_ws = q_ws + (size_t)Bn * Dn;              // 16*1024
  float* v_ws = k_ws + (size_t)Bn * KVHn * HDn;      // 16*1024
  float* a_ws = v_ws + (size_t)Bn * KVHn * HDn;      // 16*4096

  // 1) Bulk KV-cache copy (dominant traffic: 2 x 268 MB, ~46 us at 23.3 TB/s)
  const size_t kvbytes = (size_t)Bn * KVHn * Sn * HDn * sizeof(float);
  (void)hipMemcpyAsync(out_key,   past_key,   kvbytes, hipMemcpyDeviceToDevice, stream);
  (void)hipMemcpyAsync(out_value, past_value, kvbytes, hipMemcpyDeviceToDevice, stream);

  // 2) QKV projections (fp32 WMMA, M=16)
  gemm16_wmma_f32<<<Dn / 16, 32, 0, stream>>>(hidden, Wq, q_ws, Dn, Dn);
  gemm16_wmma_f32<<<(KVHn * HDn) / 16, 32, 0, stream>>>(hidden, Wk, k_ws, Dn, KVHn * HDn);
  gemm16_wmma_f32<<<(KVHn * HDn) / 16, 32, 0, stream>>>(hidden, Wv, v_ws, Dn, KVHn * HDn);

  // 3) RoPE + scatter new k/v rows into the copied caches
  rope_scatter<<<Bn * (Hn + 2 * KVHn), HDn, 0, stream>>>(
      q_ws, k_ws, v_ws, cosp, sinp, steps, out_key, out_value);

  // 4) GQA decode attention
  //    LDS: scores 64KB + combine 2KB + 4 + 2x8KB async V buffers ≈ 84KB
  const size_t smem =
      (size_t)(4 * Sn + 4 * HDn + 4 + 2 * 16 * HDn) * sizeof(float);
  attn_decode<<<Bn * KVHn, 256, smem, stream>>>(
      q_ws, k_ws, v_ws, past_key, past_value, mask, steps, a_ws);

  // 5) Output projection
  gemm16_wmma_f32<<<Dn / 16, 32, 0, stream>>>(a_ws, Wo, out_attn, Dn, Dn);
}